// Net_3633542332751
// MI455X (gfx1250) — compile-verified
//
#include <hip/hip_runtime.h>

typedef __attribute__((ext_vector_type(2))) float v2f;
typedef __attribute__((ext_vector_type(8))) float v8f;

// ---------------- problem constants (match reference) ----------------
static const int   N1 = 200000, N2 = 300000, N3 = 400000;
static const long long E1 = 3200000, E2 = 3200000, E3 = 3200000;
static const long long A2 = 600000,  A3 = 1200000;
static const int   BB = 2048, F_IN = 13, NI2 = 50, NI3 = 100, NC = 10;

#define TPB 256
static inline unsigned nblocks(long long total, int bs) {
  return (unsigned)((total + bs - 1) / bs);
}

// =====================================================================
// WMMA fp32 GEMM (fused dual-input):
//   DUAL=1:  C[M x Nout] = Aagg[M x Kpad] @ W1 + Asrc[M x Kpad] @ W2 + bias
//   DUAL=0:  C[M x Nout] = Aagg[M x Kpad] @ W1 + bias
// Requirements (guaranteed by caller):
//   M % 16 == 0, Kpad % 4 == 0, lda == Kpad, A rows zero-padded for k in
//   [Kreal,Kpad), base pointers 16B aligned, Kpad even (so b64 loads align).
// W1/W2 are Kreal x Nout row-major in global; staged zero-padded into LDS
// pair-interleaved: lw[(k>>1)*2*NoutPad + n*2 + (k&1)].
// One wave computes one 16x16 output tile with V_WMMA_F32_16X16X4_F32.
// =====================================================================
template <int DUAL>
__global__ void __launch_bounds__(256)
wmma_gemm_f32(const float* __restrict__ Aagg, const float* __restrict__ Asrc,
              const float* __restrict__ W1, const float* __restrict__ W2,
              int Kreal, int Kpad, const float* __restrict__ bias,
              float* __restrict__ Cm, int M, int Nout)
{
  extern __shared__ float smem[];
  const int NoutPad = (Nout + 15) & ~15;
  float* lw1 = smem;                       // Kpad * NoutPad, pair-interleaved
  float* lw2 = smem + Kpad * NoutPad;

  // ---- cooperative, zero-padded weight staging ----
  const int total = Kpad * NoutPad;
  for (int t = threadIdx.x; t < total; t += blockDim.x) {
    const int k = t / NoutPad;
    const int n = t - k * NoutPad;
    const int di = (k >> 1) * (2 * NoutPad) + n * 2 + (k & 1);
    const bool ok = (k < Kreal) && (n < Nout);
    lw1[di] = ok ? W1[(size_t)k * Nout + n] : 0.0f;
    if (DUAL) lw2[di] = ok ? W2[(size_t)k * Nout + n] : 0.0f;
  }
  __syncthreads();

  // ---- tile assignment (wave-uniform) ----
  const int wpb = blockDim.x >> 5;
  const long long tile = (long long)blockIdx.x * wpb + (threadIdx.x >> 5);
  const int tilesN = NoutPad >> 4;
  const int tilesM = M >> 4;
  if (tile >= (long long)tilesM * tilesN) return;
  const int tm = (int)(tile / tilesN);
  const int tn = (int)(tile % tilesN);

  const int lane = threadIdx.x & 31;
  const int half = lane >> 4;              // 0: K {0,1}/M 0..7 ; 1: K {2,3}/M 8..15
  const int l16  = lane & 15;
  const int colB = tn * 16 + l16;

  const float* a1p = Aagg + (size_t)(tm * 16 + l16) * Kpad + half * 2;
  const float* a2p = DUAL ? (Asrc + (size_t)(tm * 16 + l16) * Kpad + half * 2) : nullptr;
  const int bstep = 2 * NoutPad;           // LDS floats per pair-row
  const float* b1p = lw1 + half * bstep + colB * 2;
  const float* b2p = lw2 + half * bstep + colB * 2;

  v8f acc = {};
  for (int k0 = 0; k0 < Kpad; k0 += 4) {
    v2f a1 = *(const v2f*)(a1p + k0);                        // global_load_b64
    v2f b1 = *(const v2f*)(b1p + (k0 >> 1) * bstep);         // ds_load_b64
    acc = __builtin_amdgcn_wmma_f32_16x16x4_f32(false, a1, false, b1,
                                                (short)0, acc, false, false);
    if (DUAL) {
      v2f a2 = *(const v2f*)(a2p + k0);
      v2f b2 = *(const v2f*)(b2p + (k0 >> 1) * bstep);
      acc = __builtin_amdgcn_wmma_f32_16x16x4_f32(false, a2, false, b2,
                                                  (short)0, acc, false, false);
    }
  }

  // ---- store (only colB guard needed; M rows always valid) ----
  if (colB < Nout) {
    const float bv = bias[colB];
    float* crow = Cm + (size_t)(tm * 16 + half * 8) * Nout + colB;
    #pragma unroll
    for (int v = 0; v < 8; ++v) {
      crow[(size_t)v * Nout] = acc[v] + bv;
    }
  }
}

// ---------------- edge gather + scatter-add (float4 vectorized) -------------
// out[dst[e], 4q..4q+3] += src[src[e], 4q..4q+3], ld == F (multiple of 4)
__global__ void edge_scatter_add4(const float* __restrict__ src,
                                  const int* __restrict__ es, const int* __restrict__ ed,
                                  long long E, int F4, float* __restrict__ out)
{
  long long idx = (long long)blockIdx.x * blockDim.x + threadIdx.x;
  if (idx >= E * (long long)F4) return;
  long long e = idx / F4;
  int q = (int)(idx - e * (long long)F4);
  const float4 v = ((const float4*)(src + (size_t)es[e] * (F4 * 4)))[q];
  float* o = out + (size_t)ed[e] * (F4 * 4) + q * 4;
  atomicAdd(o + 0, v.x);
  atomicAdd(o + 1, v.y);
  atomicAdd(o + 2, v.z);
  atomicAdd(o + 3, v.w);
}

// ---------------- segment mean pieces ----------------
__global__ void seg_sum_kernel(const float* __restrict__ src, int ldsrc,
                               const int* __restrict__ gidx, const int* __restrict__ sidx,
                               long long nItems, int F,
                               float* __restrict__ out, int ldout, int coff)
{
  long long idx = (long long)blockIdx.x * blockDim.x + threadIdx.x;
  if (idx >= nItems * (long long)F) return;
  long long i = idx / F;
  int f = (int)(idx - i * (long long)F);
  int sr = gidx ? gidx[i] : (int)i;
  int sg = sidx[i];
  atomicAdd(&out[(size_t)sg * ldout + coff + f], src[(size_t)sr * ldsrc + f]);
}

__global__ void seg_cnt_kernel(const int* __restrict__ sidx, long long n,
                               float* __restrict__ cnt)
{
  long long i = (long long)blockIdx.x * blockDim.x + threadIdx.x;
  if (i >= n) return;
  atomicAdd(&cnt[sidx[i]], 1.0f);
}

__global__ void seg_div_kernel(float* __restrict__ out, int ldout, int coff,
                               const float* __restrict__ cnt, long long nSeg, int F)
{
  long long idx = (long long)blockIdx.x * blockDim.x + threadIdx.x;
  if (idx >= nSeg * (long long)F) return;
  long long s = idx / F;
  int f = (int)(idx - s * (long long)F);
  out[(size_t)s * ldout + coff + f] /= fmaxf(cnt[s], 1.0f);
}

// ---------------- misc elementwise ----------------
__global__ void elu_kernel(float* __restrict__ x, long long n)
{
  long long i = (long long)blockIdx.x * blockDim.x + threadIdx.x;
  if (i >= n) return;
  float v = x[i];
  x[i] = v > 0.0f ? v : expm1f(v);
}

__global__ void copy_cols_kernel(const float* __restrict__ src, int ldsrc,
                                 float* __restrict__ dst, int lddst, int coff,
                                 long long n, int F)
{
  long long idx = (long long)blockIdx.x * blockDim.x + threadIdx.x;
  if (idx >= n * (long long)F) return;
  long long i = idx / F;
  int f = (int)(idx - i * (long long)F);
  dst[(size_t)i * lddst + coff + f] = src[(size_t)i * ldsrc + f];
}

__global__ void log_softmax_rows(const float* __restrict__ in, float* __restrict__ out,
                                 int rows, int C)
{
  int r = blockIdx.x * blockDim.x + threadIdx.x;
  if (r >= rows) return;
  float m = -3.402823466e+38f;
  for (int c = 0; c < C; ++c) m = fmaxf(m, in[(size_t)r * C + c]);
  float s = 0.0f;
  for (int c = 0; c < C; ++c) s += expf(in[(size_t)r * C + c] - m);
  float ls = logf(s);
  for (int c = 0; c < C; ++c) out[(size_t)r * C + c] = in[(size_t)r * C + c] - m - ls;
}

// ---------------- host-side helpers (enqueue only, graph-capture safe) -------
static void run_gemm_single(const float* A, const float* W, int Kreal, int Kpad,
                            const float* bias, float* Cm, int M, int Nout,
                            hipStream_t s)
{
  const int NoutPad = (Nout + 15) & ~15;
  long long tiles = (long long)(M >> 4) * (NoutPad >> 4);
  unsigned blocks = (unsigned)((tiles + 7) / 8);
  size_t shmem = (size_t)Kpad * NoutPad * sizeof(float);
  wmma_gemm_f32<0><<<blocks, 256, shmem, s>>>(A, nullptr, W, nullptr,
                                              Kreal, Kpad, bias, Cm, M, Nout);
}

// PyG GraphConv (aggr='add'): out = segsum(src over edges)@Wr + src@Wo + br
static void graph_conv(const float* src, int N, int Kreal, int Kpad,
                       const int* ei, long long E,
                       const float* Wr, const float* br, const float* Wo,
                       int Fout, float* out, float* agg, hipStream_t s)
{
  hipMemsetAsync(agg, 0, (size_t)N * Kpad * sizeof(float), s);
  edge_scatter_add4<<<nblocks(E * (long long)(Kpad / 4), TPB), TPB, 0, s>>>(
      src, ei, ei + E, E, Kpad / 4, agg);
  const int NoutPad = (Fout + 15) & ~15;
  long long tiles = (long long)(N >> 4) * (NoutPad >> 4);
  unsigned blocks = (unsigned)((tiles + 7) / 8);
  size_t shmem = 2 * (size_t)Kpad * NoutPad * sizeof(float);
  wmma_gemm_f32<1><<<blocks, 256, shmem, s>>>(agg, src, Wr, Wo,
                                              Kreal, Kpad, br, out, N, Fout);
}

static void scatter_mean(const float* src, int ldsrc, const int* gidx,
                         const int* sidx, long long nItems, int F,
                         float* out, int ldout, int coff, int nSeg,
                         float* cnt, hipStream_t s)
{
  hipMemsetAsync(cnt, 0, (size_t)nSeg * sizeof(float), s);
  seg_sum_kernel<<<nblocks(nItems * (long long)F, TPB), TPB, 0, s>>>(
      src, ldsrc, gidx, sidx, nItems, F, out, ldout, coff);
  seg_cnt_kernel<<<nblocks(nItems, TPB), TPB, 0, s>>>(sidx, nItems, cnt);
  seg_div_kernel<<<nblocks((long long)nSeg * F, TPB), TPB, 0, s>>>(
      out, ldout, coff, cnt, nSeg, F);
}

static void elu(float* x, long long n, hipStream_t s)
{
  elu_kernel<<<nblocks(n, TPB), TPB, 0, s>>>(x, n);
}

// ---------------- entry ----------------
extern "C" void kernel_launch(void* const* d_in, const int* in_sizes, int n_in,
                              void* d_out, int out_size, void* d_ws, size_t ws_size,
                              hipStream_t stream)
{
  (void)in_sizes; (void)n_in; (void)out_size; (void)ws_size;

  const float* x    = (const float*)d_in[0];
  const int*   ei1  = (const int*)d_in[1];
  const int*   bat1 = (const int*)d_in[2];
  const int*   ai2  = (const int*)d_in[3];
  const float* iso2 = (const float*)d_in[4];
  const int*   ei2  = (const int*)d_in[5];
  const int*   bat2 = (const int*)d_in[6];
  const int*   ai3  = (const int*)d_in[7];
  const float* iso3 = (const float*)d_in[8];
  const int*   ei3  = (const int*)d_in[9];
  const int*   bat3 = (const int*)d_in[10];

  auto fp = [&](int i) { return (const float*)d_in[i]; };
  // params dict in insertion order: Wr{i}, br{i}, Wo{i} for i=1..7, then fc layers
  const float *Wr1 = fp(11), *br1 = fp(12), *Wo1 = fp(13);
  const float *Wr2 = fp(14), *br2 = fp(15), *Wo2 = fp(16);
  const float *Wr3 = fp(17), *br3 = fp(18), *Wo3 = fp(19);
  const float *Wr4 = fp(20), *br4 = fp(21), *Wo4 = fp(22);
  const float *Wr5 = fp(23), *br5 = fp(24), *Wo5 = fp(25);
  const float *Wr6 = fp(26), *br6 = fp(27), *Wo6 = fp(28);
  const float *Wr7 = fp(29), *br7 = fp(30), *Wo7 = fp(31);
  const float *Wfc1 = fp(32), *bfc1 = fp(33);
  const float *Wfc2 = fp(34), *bfc2 = fp(35);
  const float *Wfc3 = fp(36), *bfc3 = fp(37);

  // ---- carve workspace ----
  char* wsb = (char*)d_ws;
  size_t off = 0;
  auto carve = [&](size_t bytes) -> float* {
    size_t p0 = off;
    off += (bytes + 255) & ~(size_t)255;
    return (float*)(wsb + p0);
  };
  float* agg  = carve((size_t)N3 * 164 * 4);  // shared segsum accumulator (max width)
  float* xpad = carve((size_t)N1 * 16 * 4);   // x padded 13 -> 16
  float* h1a  = carve((size_t)N1 * 64 * 4);
  float* h1b  = carve((size_t)N1 * 64 * 4);
  float* h2c  = carve((size_t)N2 * 116 * 4);  // pooled(64) | iso2(50) | zero pad(2)
  float* h2a  = carve((size_t)N2 * 64 * 4);
  float* h2b  = carve((size_t)N2 * 64 * 4);
  float* h3c  = carve((size_t)N3 * 164 * 4);  // pooled(64) | iso3(100)
  float* h3a  = carve((size_t)N3 * 64 * 4);
  float* h3b  = carve((size_t)N3 * 64 * 4);
  float* cnt  = carve((size_t)N3 * 4);
  float* z    = carve((size_t)BB * 192 * 4);
  float* z1   = carve((size_t)BB * 64 * 4);
  float* z2   = carve((size_t)BB * 32 * 4);
  float* lg   = carve((size_t)BB * NC * 4);

  // ---- pad x: N1 x 13 -> N1 x 16 (zero-filled tail) ----
  hipMemsetAsync(xpad, 0, (size_t)N1 * 16 * 4, stream);
  copy_cols_kernel<<<nblocks((long long)N1 * F_IN, TPB), TPB, 0, stream>>>(
      x, F_IN, xpad, 16, 0, N1, F_IN);

  // ---- graph 1: three GraphConv + ELU ----
  graph_conv(xpad, N1, F_IN, 16, ei1, E1, Wr1, br1, Wo1, 32, h1a, agg, stream);
  elu(h1a, (long long)N1 * 32, stream);
  graph_conv(h1a, N1, 32, 32, ei1, E1, Wr2, br2, Wo2, 64, h1b, agg, stream);
  elu(h1b, (long long)N1 * 64, stream);
  graph_conv(h1b, N1, 64, 64, ei1, E1, Wr3, br3, Wo3, 64, h1a, agg, stream);
  elu(h1a, (long long)N1 * 64, stream);                     // h = h1a

  // ---- readout x1 and pooling to 2- and 3-node graphs ----
  hipMemsetAsync(z, 0, (size_t)BB * 192 * 4, stream);
  scatter_mean(h1a, 64, nullptr, bat1, N1, 64, z, 192, 0, BB, cnt, stream);

  hipMemsetAsync(h2c, 0, (size_t)N2 * 116 * 4, stream);
  scatter_mean(h1a, 64, ai2, ai2 + A2, A2, 64, h2c, 116, 0, N2, cnt, stream);
  copy_cols_kernel<<<nblocks((long long)N2 * NI2, TPB), TPB, 0, stream>>>(
      iso2, NI2, h2c, 116, 64, N2, NI2);

  hipMemsetAsync(h3c, 0, (size_t)N3 * 164 * 4, stream);
  scatter_mean(h1a, 64, ai3, ai3 + A3, A3, 64, h3c, 164, 0, N3, cnt, stream);
  copy_cols_kernel<<<nblocks((long long)N3 * NI3, TPB), TPB, 0, stream>>>(
      iso3, NI3, h3c, 164, 64, N3, NI3);

  // ---- graph 2 ----
  graph_conv(h2c, N2, 114, 116, ei2, E2, Wr4, br4, Wo4, 64, h2a, agg, stream);
  elu(h2a, (long long)N2 * 64, stream);
  graph_conv(h2a, N2, 64, 64, ei2, E2, Wr5, br5, Wo5, 64, h2b, agg, stream);
  elu(h2b, (long long)N2 * 64, stream);
  scatter_mean(h2b, 64, nullptr, bat2, N2, 64, z, 192, 64, BB, cnt, stream);

  // ---- graph 3 ----
  graph_conv(h3c, N3, 164, 164, ei3, E3, Wr6, br6, Wo6, 64, h3a, agg, stream);
  elu(h3a, (long long)N3 * 64, stream);
  graph_conv(h3a, N3, 64, 64, ei3, E3, Wr7, br7, Wo7, 64, h3b, agg, stream);
  elu(h3b, (long long)N3 * 64, stream);
  scatter_mean(h3b, 64, nullptr, bat3, N3, 64, z, 192, 128, BB, cnt, stream);

  // ---- MLP head + log_softmax ----
  run_gemm_single(z,  Wfc1, 192, 192, bfc1, z1, BB, 64, stream);
  elu(z1, (long long)BB * 64, stream);
  run_gemm_single(z1, Wfc2, 64, 64, bfc2, z2, BB, 32, stream);
  elu(z2, (long long)BB * 32, stream);
  run_gemm_single(z2, Wfc3, 32, 32, bfc3, lg, BB, NC, stream);
  log_softmax_rows<<<nblocks(BB, TPB), TPB, 0, stream>>>(lg, (float*)d_out, BB, NC);
}